// PCEN_56891136803335
// MI455X (gfx1250) — compile-verified
//
#include <hip/hip_runtime.h>
#include <stdint.h>

// PCEN for [B=32, T=4096, F=256] fp32.
// Pipeline (all on `stream`, deterministic, graph-capture safe):
//   1) ema_partial:  per (b,chunk) zero-carry EMA tail  -> ws chunk buffer
//   2) carry_fix:    compose chunk carries (EMA is linear in the carry)
//   3) minmax_init:  reset global min/max atomic slots
//   4) pcen_main:    exact EMA per chunk + pow-chain, write unnormalized out,
//                    global min/max via flipped-uint atomics
//   5) normalize:    float4 in-place global min/max normalization
// CDNA5 paths used: global_load_async_to_lds_b128 double-buffered staging with
// s_wait_asynccnt (ASYNCcnt) — the serial scan chain is decoupled from HBM
// latency; waits target only the previous tile while the next is in flight.
//
// ws layout: [0..7]  : 2 x u32 atomic slots (min,max in order-preserving space)
//            [64 ...]: B*NC*F floats (512 KB) chunk scratch.

#define EPS_ 1e-6f

constexpr int B_ = 32;
constexpr int T_ = 4096;
constexpr int F_ = 256;
constexpr int NC_ = 16;                      // chunks per sequence
constexpr int L_  = T_ / NC_;                // 256 timesteps per chunk
constexpr int TILE_T = 32;                   // timesteps per LDS tile
constexpr int NTILES = L_ / TILE_T;          // 8
constexpr int TILE_ELEMS = TILE_T * F_;      // 8192 floats = 32 KB / buffer
constexpr int LOADS_PER_THREAD = TILE_ELEMS / 4 / 256;  // 8 x b128 per thread

// Low 32 bits of a generic pointer into LDS == byte offset within wave LDS.
__device__ __forceinline__ uint32_t lds_off(const void* p) {
  return (uint32_t)(uintptr_t)p;
}

// Stage one TILE_T x F_ tile (contiguous 32 KB) into LDS asynchronously.
__device__ __forceinline__ void async_tile(const float* __restrict__ g,
                                           const float* smbuf, int tid) {
#pragma unroll
  for (int k = 0; k < LOADS_PER_THREAD; ++k) {
    const int f4 = tid + k * 256;
    const uint32_t l = lds_off(smbuf + (size_t)f4 * 4);
    const float* gp = g + (size_t)f4 * 4;
    asm volatile("global_load_async_to_lds_b128 %0, %1, off"
                 :: "v"(l), "v"(gp) : "memory");
  }
}

__device__ __forceinline__ void wait_async_prev() {
  // <= LOADS_PER_THREAD outstanding: previous tile fully landed, next in flight
  asm volatile("s_wait_asynccnt 8" ::: "memory");
}
__device__ __forceinline__ void wait_async_all() {
  asm volatile("s_wait_asynccnt 0" ::: "memory");
}

// Order-preserving float <-> uint mapping for global min/max atomics.
__device__ __forceinline__ unsigned flipf(float f) {
  unsigned u = __float_as_uint(f);
  return u ^ (unsigned)(((int)u >> 31) | 0x80000000);
}
__device__ __forceinline__ float unflipf(unsigned u) {
  unsigned m = (u & 0x80000000u) ? 0x80000000u : 0xFFFFFFFFu;
  return __uint_as_float(u ^ m);
}

// ---------------- pass 1: zero-carry chunk scan tails ----------------
__global__ void __launch_bounds__(256)
pcen_ema_partial(const float* __restrict__ x,
                 const float* __restrict__ smooth_p,
                 float* __restrict__ chunks) {
  __shared__ float sm[2][TILE_ELEMS];
  const int tid = threadIdx.x;
  const int blk = blockIdx.x;
  const int b = blk / NC_;
  const int c = blk % NC_;
  const float w = fminf(fmaxf(smooth_p[0], 0.0f), 1.0f);
  const float* gbase = x + ((size_t)b * T_ + (size_t)c * L_) * F_;

  async_tile(gbase, sm[0], tid);
  float a = 0.0f;
  for (int i = 0; i < NTILES; ++i) {
    if (i + 1 < NTILES) {
      async_tile(gbase + (size_t)(i + 1) * TILE_ELEMS, sm[(i + 1) & 1], tid);
      wait_async_prev();
    } else {
      wait_async_all();
    }
    __syncthreads();
    const float* buf = sm[i & 1];
#pragma unroll
    for (int t = 0; t < TILE_T; ++t) {
      const float xv = buf[t * F_ + tid];
      a = fmaf(w, xv - a, a);           // a = (1-w)a + w*x
    }
    __syncthreads();                    // all waves done reading before reuse
  }
  chunks[(size_t)blk * F_ + tid] = a;   // s_end for this chunk
}

// ---------------- pass 2: compose carries across chunks ----------------
__global__ void __launch_bounds__(256)
pcen_carry_fix(const float* __restrict__ x,
               const float* __restrict__ smooth_p,
               float* __restrict__ chunks) {
  const int b = blockIdx.x;
  const int f = threadIdx.x;
  const float w = fminf(fmaxf(smooth_p[0], 0.0f), 1.0f);
  const float dL = __powf(1.0f - w, (float)L_);   // (1-w)^L
  float carry = x[(size_t)b * T_ * F_ + f];       // a_{-1} = x[b,0,f]
  for (int c = 0; c < NC_; ++c) {
    const size_t idx = ((size_t)b * NC_ + c) * F_ + f;
    const float s = chunks[idx];
    chunks[idx] = carry;                          // carry_in for chunk c
    carry = fmaf(dL, carry, s);                   // carry_in for chunk c+1
  }
}

// ---------------- pass 3 init: min/max slots ----------------
__global__ void pcen_minmax_init(unsigned* __restrict__ mm) {
  mm[0] = 0xFFFFFFFFu;  // min slot (flipped space)
  mm[1] = 0u;           // max slot (flipped space)
}

// ---------------- pass 3: exact EMA + PCEN pointwise + min/max ----------------
__global__ void __launch_bounds__(256)
pcen_main(const float* __restrict__ x,
          const float* __restrict__ gain_p, const float* __restrict__ bias_p,
          const float* __restrict__ root_p, const float* __restrict__ smooth_p,
          const float* __restrict__ chunks,
          float* __restrict__ out,
          unsigned* __restrict__ mm) {
  __shared__ float sm[2][TILE_ELEMS];
  __shared__ unsigned s_mn, s_mx;
  const int tid = threadIdx.x;
  const int blk = blockIdx.x;
  const int b = blk / NC_;
  const int c = blk % NC_;

  const float w    = fminf(fmaxf(smooth_p[0], 0.0f), 1.0f);
  const float g    = fminf(gain_p[0], 1.0f);
  const float r    = fmaxf(root_p[0], 1.0f);
  const float bias = bias_p[0];
  const float oor  = 1.0f / r;
  const float br   = __powf(bias, oor);

  if (tid == 0) { s_mn = 0xFFFFFFFFu; s_mx = 0u; }

  const size_t base = ((size_t)b * T_ + (size_t)c * L_) * F_;
  const float* gbase = x + base;
  float* obase = out + base;

  float a = chunks[(size_t)blk * F_ + tid];   // exact carry into this chunk
  float vmn = __FLT_MAX__;
  float vmx = -__FLT_MAX__;

  async_tile(gbase, sm[0], tid);
  for (int i = 0; i < NTILES; ++i) {
    if (i + 1 < NTILES) {
      async_tile(gbase + (size_t)(i + 1) * TILE_ELEMS, sm[(i + 1) & 1], tid);
      wait_async_prev();
    } else {
      wait_async_all();
    }
    __syncthreads();
    const float* buf = sm[i & 1];
    float* ob = obase + (size_t)i * TILE_ELEMS;
#pragma unroll 4
    for (int t = 0; t < TILE_T; ++t) {
      const float xv = buf[t * F_ + tid];
      a = fmaf(w, xv - a, a);                       // ema at time t
      const float den = __powf(EPS_ + a, g);        // (eps+ema)^g
      const float v = __powf(xv / den + bias, oor) - br;
      ob[t * F_ + tid] = v;
      vmn = fminf(vmn, v);
      vmx = fmaxf(vmx, v);
    }
    __syncthreads();
  }

  atomicMin(&s_mn, flipf(vmn));
  atomicMax(&s_mx, flipf(vmx));
  __syncthreads();
  if (tid == 0) {
    atomicMin(&mm[0], s_mn);
    atomicMax(&mm[1], s_mx);
  }
}

// ---------------- pass 4: in-place global min/max normalize ----------------
__global__ void __launch_bounds__(256)
pcen_normalize(float4* __restrict__ out, const unsigned* __restrict__ mm) {
  const float mn = unflipf(mm[0]);
  const float mx = unflipf(mm[1]);
  const float s = 2.0f / (mx - mn);
  const float o = fmaf(-mn, s, -1.0f);
  const size_t base = (size_t)blockIdx.x * 1024 + threadIdx.x;
#pragma unroll
  for (int k = 0; k < 4; ++k) {
    const size_t i = base + (size_t)k * 256;
    float4 v = out[i];
    v.x = fmaf(v.x, s, o);
    v.y = fmaf(v.y, s, o);
    v.z = fmaf(v.z, s, o);
    v.w = fmaf(v.w, s, o);
    out[i] = v;
  }
}

extern "C" void kernel_launch(void* const* d_in, const int* in_sizes, int n_in,
                              void* d_out, int out_size, void* d_ws, size_t ws_size,
                              hipStream_t stream) {
  (void)in_sizes; (void)n_in; (void)out_size; (void)ws_size;
  const float* x        = (const float*)d_in[0];
  const float* gain_p   = (const float*)d_in[1];
  const float* bias_p   = (const float*)d_in[2];
  const float* root_p   = (const float*)d_in[3];
  const float* smooth_p = (const float*)d_in[4];
  float* out = (float*)d_out;

  unsigned* mm   = (unsigned*)d_ws;
  float* chunks  = (float*)((char*)d_ws + 64);   // B*NC*F floats = 512 KB

  pcen_ema_partial<<<B_ * NC_, 256, 0, stream>>>(x, smooth_p, chunks);
  pcen_carry_fix<<<B_, F_, 0, stream>>>(x, smooth_p, chunks);
  pcen_minmax_init<<<1, 1, 0, stream>>>(mm);
  pcen_main<<<B_ * NC_, 256, 0, stream>>>(x, gain_p, bias_p, root_p, smooth_p,
                                          chunks, out, mm);
  const int total_f4 = B_ * T_ * F_ / 4;         // 8,388,608
  pcen_normalize<<<total_f4 / 1024, 256, 0, stream>>>((float4*)out, mm);
}